// CustomGCN_74818330296407
// MI455X (gfx1250) — compile-verified
//
#include <hip/hip_runtime.h>

// ---------------------------------------------------------------------------
// CDNA5 (gfx1250) GCN: 3x (GEMM + symmetric-norm scatter-add + LeakyReLU)
//
// Roofline reasoning (MI455X):
//  * GEMM: 4.9 GFLOP total fp32 -> V_WMMA_F32_16X16X4_F32, negligible time.
//  * Edge phase: ~2 GB of gather + atomic RMW across 3 layers, BUT h/out
//    buffers are 25.6 MB each -> resident in the 192 MB L2, so the scatter is
//    L2-bandwidth bound, well under the 23.3 TB/s HBM floor (~130 us).
//    unsafeAtomicAdd -> hardware GLOBAL_ATOMIC_ADD_F32 (no return, STOREcnt).
// ---------------------------------------------------------------------------

typedef __attribute__((ext_vector_type(2))) float v2f;
typedef __attribute__((ext_vector_type(8))) float v8f;

#define DFEAT 128
#define LEAKY 0.01f

// ---------------------------------------------------------------------------
// h = x @ W^T  using V_WMMA_F32_16X16X4_F32 (wave32).
// Block = 256 threads (8 waves); each wave computes a 16x128 output panel.
// W (128x128 fp32 = 64 KB) is staged once per block into LDS.
//
// WMMA f32 16x16x4 VGPR layouts (ISA 7.12.2):
//   A (16x4):  lanes 0-15: M=lane,   v0=K0,v1=K1 ; lanes 16-31: M=lane-16, v0=K2,v1=K3
//   B (4x16):  lanes 0-15: N=lane,   v0=K0,v1=K1 ; lanes 16-31: N=lane-16, v0=K2,v1=K3
//   C/D(16x16): VGPR r: lanes 0-15 -> M=r, N=lane ; lanes 16-31 -> M=r+8, N=lane-16
// ---------------------------------------------------------------------------
__global__ void __launch_bounds__(256) gcn_gemm_xWT(
    const float* __restrict__ x, const float* __restrict__ W,
    float* __restrict__ h, int n_nodes)
{
  __shared__ float Wlds[DFEAT * DFEAT];  // 64 KB (WGP has 320 KB)

  {  // cooperative 128-bit staging of W into LDS
    const float4* Wg4 = (const float4*)W;
    float4* Wl4 = (float4*)Wlds;
    #pragma unroll
    for (int i = 0; i < (DFEAT * DFEAT / 4) / 256; ++i)
      Wl4[threadIdx.x + 256 * i] = Wg4[threadIdx.x + 256 * i];
  }
  __syncthreads();

  const int wave = threadIdx.x >> 5;
  const int lane = threadIdx.x & 31;
  const int half = lane >> 4;   // which K-pair / which M-half for C
  const int lrow = lane & 15;

  const int base = blockIdx.x * 128 + wave * 16;   // first row of this wave's panel
  if (base >= n_nodes) return;                     // wave-uniform: EXEC stays all-1 for WMMA

  int row = base + lrow;
  if (row >= n_nodes) row = n_nodes - 1;           // clamp loads; stores are guarded
  const v2f* xr = (const v2f*)(x + (size_t)row * DFEAT);

  // A fragments for the full K=128 (32 ktiles): 64 VGPRs
  v2f afrag[32];
  #pragma unroll
  for (int kt = 0; kt < 32; ++kt) afrag[kt] = xr[2 * kt + half];

  #pragma unroll
  for (int nt = 0; nt < 8; ++nt) {
    // B[k][n] = W[n][k]; lane owns output column n = nt*16 + lrow
    const v2f* wr = (const v2f*)(Wlds + (nt * 16 + lrow) * DFEAT);
    v8f acc = {};
    #pragma unroll
    for (int kt = 0; kt < 32; ++kt) {
      v2f b = wr[2 * kt + half];
      acc = __builtin_amdgcn_wmma_f32_16x16x4_f32(
          /*neg_a=*/false, afrag[kt], /*neg_b=*/false, b,
          /*c_mod=*/(short)0, acc, /*reuse_a=*/false, /*reuse_b=*/false);
    }
    const int col = nt * 16 + lrow;
    #pragma unroll
    for (int r = 0; r < 8; ++r) {
      int m = base + half * 8 + r;
      if (m < n_nodes) h[(size_t)m * DFEAT + col] = acc[r];
    }
  }
}

// --------------------------- degree / normalization ------------------------
__global__ void deg_init(float* __restrict__ deg, int n) {
  int i = blockIdx.x * blockDim.x + threadIdx.x;
  if (i < n) deg[i] = 1.0f;  // self-loop contributes 1 to every node
}
__global__ void deg_accum(const int* __restrict__ dst, float* __restrict__ deg, int E) {
  int e = blockIdx.x * blockDim.x + threadIdx.x;
  if (e < E) unsafeAtomicAdd(&deg[dst[e]], 1.0f);  // global_atomic_add_f32
}
__global__ void deg_finalize(float* __restrict__ deg, int n) {
  int i = blockIdx.x * blockDim.x + threadIdx.x;
  if (i < n) deg[i] = __frsqrt_rn(deg[i]);  // deg >= 1 always (self-loop)
}

// out[i][f] = b[f] + h[i][f] * dis[i]^2   (bias + self-loop message, fused)
__global__ void gcn_init_out(const float* __restrict__ h, const float* __restrict__ dis,
                             const float* __restrict__ bias, float* __restrict__ out, int n) {
  int idx = blockIdx.x * blockDim.x + threadIdx.x;
  if (idx >= n * DFEAT) return;
  int i = idx >> 7, f = idx & (DFEAT - 1);
  float dv = dis[i];
  out[idx] = bias[f] + h[idx] * dv * dv;
}

// one wave per edge: out[dst][:] += h[src][:] * dis[src]*dis[dst]
__global__ void __launch_bounds__(256) gcn_scatter(
    const int* __restrict__ src, const int* __restrict__ dst,
    const float* __restrict__ h, const float* __restrict__ dis,
    float* __restrict__ out, int E)
{
  int e = blockIdx.x * 8 + (threadIdx.x >> 5);
  if (e >= E) return;
  int lane = threadIdx.x & 31;
  int s = src[e], d = dst[e];
  float nrm = dis[s] * dis[d];
  const float* hs = h + (size_t)s * DFEAT;
  float* od = out + (size_t)d * DFEAT;
  #pragma unroll
  for (int j = 0; j < 4; ++j) {
    int f = lane + 32 * j;
    unsafeAtomicAdd(&od[f], hs[f] * nrm);  // global_atomic_add_f32, no return
  }
}

__global__ void gcn_leaky(const float* __restrict__ in, float* __restrict__ out, int n) {
  int idx = blockIdx.x * blockDim.x + threadIdx.x;
  if (idx < n) {
    float v = in[idx];
    out[idx] = (v >= 0.0f) ? v : LEAKY * v;
  }
}

// ---------------------------------------------------------------------------
extern "C" void kernel_launch(void* const* d_in, const int* in_sizes, int n_in,
                              void* d_out, int out_size, void* d_ws, size_t ws_size,
                              hipStream_t stream) {
  const float* x  = (const float*)d_in[0];   // [N,128] fp32
  const int*   ei = (const int*)d_in[1];     // [2,E] int
  const float* Ws = (const float*)d_in[2];   // [3,128,128]
  const float* bs = (const float*)d_in[3];   // [3,128]
  float* out = (float*)d_out;                // [N,128]

  const int n = in_sizes[0] / DFEAT;
  const int E = in_sizes[1] / 2;
  const int* src = ei;
  const int* dst = ei + E;

  // workspace layout: xbuf [N*128] | hbuf [N*128] | dis [N]  (~51.4 MB)
  float* xbuf = (float*)d_ws;
  float* hbuf = xbuf + (size_t)n * DFEAT;
  float* dis  = hbuf + (size_t)n * DFEAT;

  const int TB = 256;
  const int nd = n * DFEAT;

  deg_init    <<<(n + TB - 1) / TB, TB, 0, stream>>>(dis, n);
  deg_accum   <<<(E + TB - 1) / TB, TB, 0, stream>>>(dst, dis, E);
  deg_finalize<<<(n + TB - 1) / TB, TB, 0, stream>>>(dis, n);

  const float* cur = x;
  for (int l = 0; l < 3; ++l) {
    gcn_gemm_xWT<<<(n + 127) / 128, 256, 0, stream>>>(cur, Ws + (size_t)l * DFEAT * DFEAT, hbuf, n);
    gcn_init_out<<<(nd + TB - 1) / TB, TB, 0, stream>>>(hbuf, dis, bs + (size_t)l * DFEAT, out, n);
    gcn_scatter <<<(E + 7) / 8, 256, 0, stream>>>(src, dst, hbuf, dis, out, E);
    float* act_dst = (l == 2) ? out : xbuf;  // last layer activates in place in d_out
    gcn_leaky   <<<(nd + TB - 1) / TB, TB, 0, stream>>>(out, act_dst, nd);
    cur = xbuf;
  }
}